// SimpleLSHAttention16_15650860826846
// MI455X (gfx1250) — compile-verified
//
#include <hip/hip_runtime.h>
#include <hip/hip_bf16.h>
#include <stdint.h>

typedef __attribute__((ext_vector_type(2))) float v2f;
typedef __attribute__((ext_vector_type(8))) float v8f;

#define BH    32          // B*H
#define SEQ   2048        // S
#define DIM   64          // D
#define DE    65          // D+1
#define DP    68          // padded K-dim: 17 * 4
#define KSEL  32          // bucket_size
#define ROWS  16          // rows per block (WMMA M)
#define NWAVE 8           // waves per block (256 threads, wave32)

__device__ __forceinline__ float wave_sum(float v) {
#pragma unroll
  for (int o = 16; o > 0; o >>= 1) v += __shfl_xor(v, o, 32);
  return v;
}

// Kernel 1: per column j build K_j = qk_ext_j * Q_j (padded to 68, zeroed on NaN).
// One wave per column: lane handles elements {lane, lane+32}; wave32 shuffle reduce.
__global__ void lsh_precompute_K(const float* __restrict__ qk,
                                 const float* __restrict__ a,
                                 float* __restrict__ Kw) {
  const int lane = threadIdx.x & 31;
  const int wave = threadIdx.x >> 5;
  const int col  = blockIdx.x * NWAVE + wave;          // [0, BH*SEQ)
  const float* q  = qk + (size_t)col * DIM;
  const float* av = a  + (size_t)col * DE;

  float q0 = q[lane], q1 = q[lane + 32];
  float nrm = sqrtf(wave_sum(q0 * q0 + q1 * q1));
  float h0 = q0 / nrm, h1 = q1 / nrm;
  float h  = sqrtf(wave_sum(h0 * h0 + h1 * h1));
  float cst = sqrtf(1.0f - h * h);                      // NaN if roundoff < 0 (faithful)

  float a0 = av[lane], a1 = av[lane + 32];
  float Q = wave_sum(q0 * a0 + q1 * a1) + cst * av[DIM];

  float* kp = Kw + (size_t)col * DP;
  if (Q != Q) {  // NaN column -> scores exactly 0 (matches where(isnan, 0))
    kp[lane] = 0.0f; kp[lane + 32] = 0.0f;
    if (lane == 0) { kp[64] = 0.0f; kp[65] = 0.0f; kp[66] = 0.0f; kp[67] = 0.0f; }
  } else {
    kp[lane] = q0 * Q; kp[lane + 32] = q1 * Q;
    if (lane == 0) { kp[64] = cst * Q; kp[65] = 0.0f; kp[66] = 0.0f; kp[67] = 0.0f; }
  }
}

// Kernel 2: one block per (bh, 16-row tile). fp32 WMMA scores -> LDS -> top-32 mask.
__global__ void lsh_scores_topk(const float* __restrict__ a,
                                const float* __restrict__ Kw,
                                float* __restrict__ out) {
  __shared__ float sc[ROWS * SEQ];     // 128 KB score tile (LDS = 320KB/WGP on CDNA5)
  __shared__ float arow[ROWS * DP];    // 4.25 KB padded A rows

  const int lane = threadIdx.x & 31;
  const int wave = threadIdx.x >> 5;
  const int bh = blockIdx.x >> 7;      // SEQ/ROWS = 128 row-blocks per head
  const int rb = blockIdx.x & 127;
  const int i0 = rb * ROWS;

  // Stage 16 rows of `a`, zero-padded 65 -> 68.
  for (int idx = threadIdx.x; idx < ROWS * DP; idx += blockDim.x) {
    int r = idx / DP, d = idx - r * DP;
    arow[idx] = (d < DE) ? a[((size_t)bh * SEQ + i0 + r) * DE + d] : 0.0f;
  }
  __syncthreads();

  const int l15  = lane & 15;
  const int half = (lane < 16) ? 0 : 2;   // ISA 16x4 f32 A: VGPR0 k in {0,2}, VGPR1 {1,3}

  // Phase A: each wave owns column tiles tile, tile+8, ... (16 tiles of 16 cols).
  for (int tile = wave; tile < SEQ / 16; tile += NWAVE) {
    const int n0 = tile * 16;
    const float* kcol = Kw + ((size_t)bh * SEQ + n0 + l15) * DP;   // L2-resident
    if (tile + NWAVE < SEQ / 16)       // gfx1250 global_prefetch for next K tile
      __builtin_prefetch(kcol + (size_t)16 * NWAVE * DP, 0, 3);
    v8f acc = {};
#pragma unroll
    for (int kk = 0; kk < DP / 4; ++kk) {
      v2f av_, bv;
      av_.x = arow[l15 * DP + kk * 4 + half];       // A: m = lane&15
      av_.y = arow[l15 * DP + kk * 4 + half + 1];
      bv.x  = kcol[kk * 4 + half];                  // B: n = lane&15, k per VGPR/half
      bv.y  = kcol[kk * 4 + half + 1];
      acc = __builtin_amdgcn_wmma_f32_16x16x4_f32(false, av_, false, bv,
                                                  (short)0, acc, false, false);
    }
#pragma unroll
    for (int r = 0; r < 8; ++r)                     // C/D: VGPR r -> M = r (+8 hi half)
      sc[(r + (half ? 8 : 0)) * SEQ + n0 + l15] = acc[r];
  }
  __syncthreads();

  // Phase B: per-row top-32 with jax.lax.top_k tie semantics (lowest index wins).
  // u32 monotonic keys only (64 VGPR/lane): find 32nd-largest value T by walking
  // distinct maxima, then rank equal values in ascending-j order via ballot/popc.
  const size_t outbase = ((size_t)bh * SEQ + i0) * (size_t)SEQ;
  for (int rr = 0; rr < ROWS / NWAVE; ++rr) {
    const int row = wave * (ROWS / NWAVE) + rr;
    uint32_t val[SEQ / 32];
#pragma unroll
    for (int t = 0; t < SEQ / 32; ++t) {
      uint32_t u = __float_as_uint(sc[row * SEQ + lane + 32 * t]);
      val[t] = (u & 0x80000000u) ? ~u : (u | 0x80000000u);   // monotonic order map
    }
    uint32_t prev = 0u, T = 0u;
    int total = 0, need_eq = 0;
    bool first = true;
    for (int it = 0; it < KSEL; ++it) {   // each step consumes >=1 distinct value
      uint32_t m = 0u;
#pragma unroll
      for (int t = 0; t < SEQ / 32; ++t) {
        uint32_t v = val[t];
        if ((first || v < prev) && v > m) m = v;   // max strictly below prev pick
      }
#pragma unroll
      for (int o = 16; o > 0; o >>= 1) {
        uint32_t other = __shfl_xor(m, o, 32);
        if (other > m) m = other;
      }
      int c = 0;
#pragma unroll
      for (int t = 0; t < SEQ / 32; ++t) c += (val[t] == m);
#pragma unroll
      for (int o = 16; o > 0; o >>= 1) c += __shfl_xor(c, o, 32);
      if (total + c >= KSEL) { T = m; need_eq = KSEL - total; break; }
      total += c;
      prev = m;
      first = false;
    }
    // Mark: v > T always selected; v == T selected for first need_eq in j order.
    // j = lane + 32*t, so ascending t (outer), ascending lane (ballot prefix) == ascending j.
    int eqseen = 0;
#pragma unroll 4
    for (int t = 0; t < SEQ / 32; ++t) {
      uint32_t v = val[t];
      uint32_t em = (uint32_t)__ballot(v == T);
      int rank = eqseen + __popc(em & ((1u << lane) - 1u));
      bool sel = (v > T) || ((v == T) && (rank < need_eq));
      __builtin_nontemporal_store(sel ? 0.0f : -10000.0f,
          &out[outbase + (size_t)row * SEQ + lane + 32 * t]);  // NT: keep K in L2
      eqseen += __popc(em);
    }
  }
}

extern "C" void kernel_launch(void* const* d_in, const int* in_sizes, int n_in,
                              void* d_out, int out_size, void* d_ws, size_t ws_size,
                              hipStream_t stream) {
  const float* qk = (const float*)d_in[0];   // [2,16,2048,64] fp32
  const float* a  = (const float*)d_in[1];   // [2,16,2048,65] fp32
  float* out = (float*)d_out;                // [2,16,2048,2048] fp32
  float* Kw  = (float*)d_ws;                 // 32*2048*68 floats = 17.8 MB (fits L2)

  lsh_precompute_K<<<BH * SEQ / NWAVE, 32 * NWAVE, 0, stream>>>(qk, a, Kw);
  lsh_scores_topk<<<BH * (SEQ / ROWS), 32 * NWAVE, 0, stream>>>(a, Kw, out);
}